// GAT_Net_89300960018825
// MI455X (gfx1250) — compile-verified
//
#include <hip/hip_runtime.h>
#include <math.h>

typedef float v2f __attribute__((ext_vector_type(2)));
typedef float v8f __attribute__((ext_vector_type(8)));

#define WAVE 32
#define GEMM_WAVES 8
#define NEG_SLOPE 0.2f

// ---------------------------------------------------------------- utilities
__global__ __launch_bounds__(256) void fill_kernel(float* __restrict__ p, float v, int n) {
    int i = blockIdx.x * blockDim.x + threadIdx.x;
    if (i < n) p[i] = v;
}

__device__ inline void atomicMaxF(float* addr, float val) {
    // Works for mixed-sign floats: positive floats ordered as ints,
    // negative floats reverse-ordered as unsigned.
    if (val >= 0.0f) atomicMax((int*)addr, __float_as_int(val));
    else             atomicMin((unsigned int*)addr, __float_as_uint(val));
}

// ---------------------------------------------------------------- dual GEMM
// outL = X @ Wl + bl ; outR = X @ Wr + br    (X:[N x K], W:[K x 64], out:[N x 64])
// One wave handles one 16-row tile; 4 col-tiles x 2 weight matrices = 8 accumulators.
__global__ __launch_bounds__(256) void gemm_dual_kernel(
    const float* __restrict__ X,
    const float* __restrict__ Wl, const float* __restrict__ bl,
    const float* __restrict__ Wr, const float* __restrict__ br,
    float* __restrict__ outL, float* __restrict__ outR,
    int N, int K)
{
    const int D = 64;
    int wave = threadIdx.x / WAVE;
    int lane = threadIdx.x % WAVE;
    int numTiles = (N + 15) / 16;
    int tile = blockIdx.x * GEMM_WAVES + wave;
    if (tile >= numTiles) return;

    int rowBase = tile * 16;
    int half = lane >> 4;   // 0: lanes 0-15, 1: lanes 16-31
    int l16  = lane & 15;

#if defined(__gfx1250__) && __has_builtin(__builtin_amdgcn_wmma_f32_16x16x4_f32)
    // A 16x4 layout: VGPR0 = K=k+0 (lanes 0-15) / K=k+2 (lanes 16-31); VGPR1 = k+1 / k+3
    int arow = rowBase + l16; if (arow >= N) arow = N - 1;   // clamp; store is guarded
    v8f accL0 = {}, accL1 = {}, accL2 = {}, accL3 = {};
    v8f accR0 = {}, accR1 = {}, accR2 = {}, accR3 = {};
    for (int kk = 0; kk < K; kk += 4) {
        const float* ap = X + (size_t)arow * K + kk + 2 * half;
        v2f a; a.x = ap[0]; a.y = ap[1];
        const float* wl0 = Wl + (size_t)(kk + 2 * half) * D + l16;
        const float* wr0 = Wr + (size_t)(kk + 2 * half) * D + l16;
        #pragma unroll
        for (int t = 0; t < 4; ++t) {
            v2f bL, bR;
            bL.x = wl0[t * 16];  bL.y = wl0[t * 16 + D];
            bR.x = wr0[t * 16];  bR.y = wr0[t * 16 + D];
            v8f* aL = (t == 0) ? &accL0 : (t == 1) ? &accL1 : (t == 2) ? &accL2 : &accL3;
            v8f* aR = (t == 0) ? &accR0 : (t == 1) ? &accR1 : (t == 2) ? &accR2 : &accR3;
            *aL = __builtin_amdgcn_wmma_f32_16x16x4_f32(false, a, false, bL, (short)0, *aL, false, false);
            *aR = __builtin_amdgcn_wmma_f32_16x16x4_f32(false, a, false, bR, (short)0, *aR, false, false);
        }
    }
    // C/D 16x16 layout: VGPR i holds row (i + 8*half), col = l16
    #pragma unroll
    for (int t = 0; t < 4; ++t) {
        int col = t * 16 + l16;
        float cbl = bl[col], cbr = br[col];
        const v8f* aL = (t == 0) ? &accL0 : (t == 1) ? &accL1 : (t == 2) ? &accL2 : &accL3;
        const v8f* aR = (t == 0) ? &accR0 : (t == 1) ? &accR1 : (t == 2) ? &accR2 : &accR3;
        #pragma unroll
        for (int i = 0; i < 8; ++i) {
            int row = rowBase + half * 8 + i;
            if (row < N) {
                outL[(size_t)row * D + col] = (*aL)[i] + cbl;
                outR[(size_t)row * D + col] = (*aR)[i] + cbr;
            }
        }
    }
#else
    // Scalar fallback (host pass / builtin missing)
    for (int idx = lane; idx < 16 * D; idx += WAVE) {
        int r = rowBase + idx / D, c = idx % D;
        if (r >= N) continue;
        float sL = bl[c], sR = br[c];
        for (int k = 0; k < K; ++k) {
            float xv = X[(size_t)r * K + k];
            sL += xv * Wl[(size_t)k * D + c];
            sR += xv * Wr[(size_t)k * D + c];
        }
        outL[(size_t)r * D + c] = sL;
        outR[(size_t)r * D + c] = sR;
    }
#endif
}

// ---------------------------------------------------------------- edge pass 1
// e[edge,h] = att[h,:] . leaky_relu(xl[src,h,:] + xr[dst,h,:]); segment-max into nmax[dst,h]
__global__ __launch_bounds__(256) void edge_score_kernel(
    const float* __restrict__ xl, const float* __restrict__ xr,
    const float* __restrict__ att, const int* __restrict__ ei,
    int E, int EE, int H, int C,
    float* __restrict__ ebuf, float* __restrict__ nmax)
{
    int idx = blockIdx.x * blockDim.x + threadIdx.x;
    if (idx >= EE * H) return;
    int e = idx / H, h = idx - e * H;
    int src, dst;
    if (e < E) { src = ei[e]; dst = ei[E + e]; } else { src = dst = e - E; }
    const float* pl = xl + (size_t)src * 64 + h * C;
    const float* pr = xr + (size_t)dst * 64 + h * C;
    const float* pa = att + h * C;
    float sum = 0.0f;
    for (int c = 0; c < C; c += 4) {
        float4 a = *(const float4*)(pl + c);
        float4 b = *(const float4*)(pr + c);
        float4 w = *(const float4*)(pa + c);
        float v0 = a.x + b.x; v0 = v0 > 0.0f ? v0 : NEG_SLOPE * v0;
        float v1 = a.y + b.y; v1 = v1 > 0.0f ? v1 : NEG_SLOPE * v1;
        float v2 = a.z + b.z; v2 = v2 > 0.0f ? v2 : NEG_SLOPE * v2;
        float v3 = a.w + b.w; v3 = v3 > 0.0f ? v3 : NEG_SLOPE * v3;
        sum += w.x * v0 + w.y * v1 + w.z * v2 + w.w * v3;
    }
    ebuf[(size_t)e * H + h] = sum;
    atomicMaxF(&nmax[(size_t)dst * H + h], sum);
}

// ---------------------------------------------------------------- edge pass 2
// a = exp(e - nmax[dst]); segment-sum into denom[dst]
__global__ __launch_bounds__(256) void edge_exp_kernel(
    const int* __restrict__ ei, int E, int EE, int H,
    float* __restrict__ ebuf, const float* __restrict__ nmax,
    float* __restrict__ denom)
{
    int idx = blockIdx.x * blockDim.x + threadIdx.x;
    if (idx >= EE * H) return;
    int e = idx / H, h = idx - e * H;
    int dst = (e < E) ? ei[E + e] : (e - E);
    float a = expf(ebuf[(size_t)e * H + h] - nmax[(size_t)dst * H + h]);
    ebuf[(size_t)e * H + h] = a;
    atomicAdd(&denom[(size_t)dst * H + h], a);
}

// ---------------------------------------------------------------- edge pass 3
// out[dst,:] += (a / denom[dst,h]) * xl[src,:]   (thread = edge x float4-chunk)
__global__ __launch_bounds__(256) void edge_scatter_kernel(
    const float* __restrict__ xl, const int* __restrict__ ei,
    int E, int EE, int H, int C,
    const float* __restrict__ ebuf, const float* __restrict__ denom,
    float* __restrict__ out)
{
    const int Q = 16;  // 64 / 4
    int idx = blockIdx.x * blockDim.x + threadIdx.x;
    if (idx >= EE * Q) return;
    int e = idx / Q, q = idx - e * Q;
    int src, dst;
    if (e < E) { src = ei[e]; dst = ei[E + e]; } else { src = dst = e - E; }
    int h = (q * 4) / C;
    float alpha = ebuf[(size_t)e * H + h] / denom[(size_t)dst * H + h];
    float4 v = *(const float4*)(xl + (size_t)src * 64 + q * 4);
    float* po = out + (size_t)dst * 64 + q * 4;
    atomicAdd(po + 0, alpha * v.x);
    atomicAdd(po + 1, alpha * v.y);
    atomicAdd(po + 2, alpha * v.z);
    atomicAdd(po + 3, alpha * v.w);
}

// ---------------------------------------------------------------- elementwise
__global__ __launch_bounds__(256) void elu_bias_kernel(
    const float* __restrict__ in, const float* __restrict__ bias,
    float* __restrict__ out, int n)
{
    int i = blockIdx.x * blockDim.x + threadIdx.x;
    if (i >= n) return;
    float v = in[i] + bias[i & 63];
    out[i] = v > 0.0f ? v : (expf(v) - 1.0f);
}

// y[n] = (h2[n,:] + bias2) . Wlin + blin
__global__ __launch_bounds__(256) void final_linear_kernel(
    const float* __restrict__ h2, const float* __restrict__ bias2,
    const float* __restrict__ Wlin, const float* __restrict__ blin,
    float* __restrict__ y, int N)
{
    int n = blockIdx.x * blockDim.x + threadIdx.x;
    if (n >= N) return;
    const float* p = h2 + (size_t)n * 64;
    float s = 0.0f;
    #pragma unroll
    for (int d = 0; d < 64; ++d) s += (p[d] + bias2[d]) * Wlin[d];
    y[n] = s + blin[0];
}

// ---------------------------------------------------------------- launch
extern "C" void kernel_launch(void* const* d_in, const int* in_sizes, int n_in,
                              void* d_out, int out_size, void* d_ws, size_t ws_size,
                              hipStream_t stream) {
    const float* x     = (const float*)d_in[0];
    const int*   ei    = (const int*)  d_in[1];
    const float* W1l   = (const float*)d_in[2];
    const float* b1l   = (const float*)d_in[3];
    const float* W1r   = (const float*)d_in[4];
    const float* b1r   = (const float*)d_in[5];
    const float* att1  = (const float*)d_in[6];
    const float* bias1 = (const float*)d_in[7];
    const float* W2l   = (const float*)d_in[8];
    const float* b2l   = (const float*)d_in[9];
    const float* W2r   = (const float*)d_in[10];
    const float* b2r   = (const float*)d_in[11];
    const float* att2  = (const float*)d_in[12];
    const float* bias2 = (const float*)d_in[13];
    const float* Wlin  = (const float*)d_in[14];
    const float* blin  = (const float*)d_in[15];

    const int Fin = 128, D = 64;
    int N  = in_sizes[0] / Fin;
    int E  = in_sizes[1] / 2;
    int EE = E + N;   // self loops appended

    // Workspace layout (buffers recycled between layers)
    float* ws    = (float*)d_ws;
    size_t off = 0;
    float* xl    = ws + off; off += (size_t)N * D;
    float* xr    = ws + off; off += (size_t)N * D;
    float* acc   = ws + off; off += (size_t)N * D;   // GAT accumulation
    float* h1    = ws + off; off += (size_t)N * D;   // elu output
    float* ebuf  = ws + off; off += (size_t)EE * 4;  // max H = 4
    float* nmax  = ws + off; off += (size_t)N * 4;
    float* denom = ws + off; off += (size_t)N * 4;
    (void)off; (void)ws_size;

    auto cdiv = [](int a, int b) { return (a + b - 1) / b; };
    const int B = 256;
    int tiles = cdiv(N, 16);
    int gemmGrid = cdiv(tiles, GEMM_WAVES);

    // ---------------- layer 1: H=4, C=16 ----------------
    fill_kernel<<<cdiv(N * 4, B), B, 0, stream>>>(nmax, -INFINITY, N * 4);
    fill_kernel<<<cdiv(N * 4, B), B, 0, stream>>>(denom, 0.0f, N * 4);
    fill_kernel<<<cdiv(N * D, B), B, 0, stream>>>(acc, 0.0f, N * D);

    gemm_dual_kernel<<<gemmGrid, B, 0, stream>>>(x, W1l, b1l, W1r, b1r, xl, xr, N, Fin);
    edge_score_kernel<<<cdiv(EE * 4, B), B, 0, stream>>>(xl, xr, att1, ei, E, EE, 4, 16, ebuf, nmax);
    edge_exp_kernel<<<cdiv(EE * 4, B), B, 0, stream>>>(ei, E, EE, 4, ebuf, nmax, denom);
    edge_scatter_kernel<<<cdiv(EE * 16, B), B, 0, stream>>>(xl, ei, E, EE, 4, 16, ebuf, denom, acc);
    elu_bias_kernel<<<cdiv(N * D, B), B, 0, stream>>>(acc, bias1, h1, N * D);

    // ---------------- layer 2: H=1, C=64 ----------------
    fill_kernel<<<cdiv(N, B), B, 0, stream>>>(nmax, -INFINITY, N);
    fill_kernel<<<cdiv(N, B), B, 0, stream>>>(denom, 0.0f, N);
    fill_kernel<<<cdiv(N * D, B), B, 0, stream>>>(acc, 0.0f, N * D);

    gemm_dual_kernel<<<gemmGrid, B, 0, stream>>>(h1, W2l, b2l, W2r, b2r, xl, xr, N, D);
    edge_score_kernel<<<cdiv(EE, B), B, 0, stream>>>(xl, xr, att2, ei, E, EE, 1, 64, ebuf, nmax);
    edge_exp_kernel<<<cdiv(EE, B), B, 0, stream>>>(ei, E, EE, 1, ebuf, nmax, denom);
    edge_scatter_kernel<<<cdiv(EE * 16, B), B, 0, stream>>>(xl, ei, E, EE, 1, 64, ebuf, denom, acc);

    final_linear_kernel<<<cdiv(N, B), B, 0, stream>>>(acc, bias2, Wlin, blin, (float*)d_out, N);
}